// Structural_Extractor_74062416053230
// MI455X (gfx1250) — compile-verified
//
#include <hip/hip_runtime.h>
#include <math.h>

// Problem constants (from reference setup_inputs)
#define B_DIM 32
#define F_DIM 65536
#define K_NBR 3
#define CH    64

typedef __attribute__((ext_vector_type(2))) float v2f;
typedef __attribute__((ext_vector_type(8))) float v8f;

// feat[b,f,:] = min_k ( l2norm(nbr_k) * l2norm(nf) )   (elementwise, per component)
__device__ __forceinline__ void compute_feat(const float* __restrict__ normals,
                                             const int* __restrict__ ring,
                                             int b, int f,
                                             float& o0, float& o1, float& o2) {
  const float* nb = normals + (size_t)b * 3 * F_DIM;   // [3, F] for this batch
  float cx = nb[f];
  float cy = nb[F_DIM + f];
  float cz = nb[2 * F_DIM + f];
  float ci = 1.0f / fmaxf(sqrtf(cx * cx + cy * cy + cz * cz), 1e-12f);
  cx *= ci; cy *= ci; cz *= ci;
  const int* rp = ring + ((size_t)b * F_DIM + f) * K_NBR;
  float m0 = 3.402823466e38f, m1 = m0, m2 = m0;
#pragma unroll
  for (int k = 0; k < K_NBR; ++k) {
    int j = rp[k];
    float x = nb[j], y = nb[F_DIM + j], z = nb[2 * F_DIM + j];
    float inv = 1.0f / fmaxf(sqrtf(x * x + y * y + z * z), 1e-12f);
    m0 = fminf(m0, x * inv * cx);
    m1 = fminf(m1, y * inv * cy);
    m2 = fminf(m2, z * inv * cz);
  }
  o0 = m0; o1 = m1; o2 = m2;
}

// ---------- Pass 1: accumulate 9 moments of feat (E[f], E[f f^T]) ----------
__global__ void moments_kernel(const float* __restrict__ normals,
                               const int* __restrict__ ring,
                               float* __restrict__ mom) {
  __shared__ float sm[9];
  if (threadIdx.x < 9) sm[threadIdx.x] = 0.0f;
  __syncthreads();

  float a[9];
#pragma unroll
  for (int i = 0; i < 9; ++i) a[i] = 0.0f;

  const long long total = (long long)B_DIM * F_DIM;
  const long long stride = (long long)gridDim.x * blockDim.x;
  for (long long i = (long long)blockIdx.x * blockDim.x + threadIdx.x; i < total; i += stride) {
    int b = (int)(i >> 16);          // F_DIM == 1<<16
    int f = (int)(i & (F_DIM - 1));
    float f0, f1, f2;
    compute_feat(normals, ring, b, f, f0, f1, f2);
    a[0] += f0;       a[1] += f1;       a[2] += f2;
    a[3] += f0 * f0;  a[4] += f0 * f1;  a[5] += f0 * f2;
    a[6] += f1 * f1;  a[7] += f1 * f2;  a[8] += f2 * f2;
  }
#pragma unroll
  for (int i = 0; i < 9; ++i) atomicAdd(&sm[i], a[i]);
  __syncthreads();
  if (threadIdx.x < 9) atomicAdd(&mom[threadIdx.x], sm[threadIdx.x]);
}

// ---------- Fold BN into conv: W' = s*W (K padded to 4 with 0), b' ----------
__global__ void fold_kernel(const float* __restrict__ conv_w,  // [64,3]
                            const float* __restrict__ conv_b,  // [64]
                            const float* __restrict__ gamma,
                            const float* __restrict__ beta,
                            const float* __restrict__ mom,
                            float* __restrict__ Wf,            // [64,4]
                            float* __restrict__ bf) {          // [64]
  int o = threadIdx.x;  // 64 threads
  const float invN = 1.0f / (float)((long long)B_DIM * F_DIM);
  float m0 = mom[0] * invN, m1 = mom[1] * invN, m2 = mom[2] * invN;
  float c00 = mom[3] * invN - m0 * m0;
  float c01 = mom[4] * invN - m0 * m1;
  float c02 = mom[5] * invN - m0 * m2;
  float c11 = mom[6] * invN - m1 * m1;
  float c12 = mom[7] * invN - m1 * m2;
  float c22 = mom[8] * invN - m2 * m2;

  float w0 = conv_w[o * 3 + 0], w1 = conv_w[o * 3 + 1], w2 = conv_w[o * 3 + 2];
  float var = w0 * w0 * c00 + w1 * w1 * c11 + w2 * w2 * c22
            + 2.0f * (w0 * w1 * c01 + w0 * w2 * c02 + w1 * w2 * c12);
  var = fmaxf(var, 0.0f);
  float mean_y = w0 * m0 + w1 * m1 + w2 * m2 + conv_b[o];
  float s = gamma[o] * rsqrtf(var + 1e-5f);

  Wf[o * 4 + 0] = s * w0;
  Wf[o * 4 + 1] = s * w1;
  Wf[o * 4 + 2] = s * w2;
  Wf[o * 4 + 3] = 0.0f;                       // K=3 -> 4 zero pad
  bf[o] = beta[o] + s * (conv_b[o] - mean_y); // out = s*y + (beta - s*mean_y)
}

// ---------- Pass 2: feat -> folded conv via WMMA f32 16x16x4 -> ReLU -> NT store ----
// One wave handles 16 consecutive f positions. Lanes 0-15 and 16-31 both compute
// feat for position (lane&15), which directly materializes the B operand layout:
//   B.v0: lanes 0-15 = K0 row, lanes 16-31 = K2 row;  B.v1: K1 row / K3(=0) row.
// A per 16-channel group: lanes 0-15 hold W'[m][0],W'[m][1]; lanes 16-31 W'[m][2],0.
// C carries the folded bias broadcast along columns. 4 WMMAs cover all 64 channels.
__global__ void __launch_bounds__(256)
conv_bn_relu_kernel(const float* __restrict__ normals,
                    const int* __restrict__ ring,
                    const float* __restrict__ Wf,
                    const float* __restrict__ bf,
                    float* __restrict__ out) {
  const int lane = threadIdx.x & 31;
  const int wave = threadIdx.x >> 5;
  const int tile = blockIdx.x * 8 + wave;          // 16 positions per tile
  const long long base = (long long)tile * 16;
  const int b     = (int)(base >> 16);             // F_DIM == 1<<16, tiles never span b
  const int fbase = (int)(base & (F_DIM - 1));
  const int m  = lane & 15;
  const bool hi = lane >= 16;
  const int p = fbase + m;

  float f0, f1, f2;
  compute_feat(normals, ring, b, p, f0, f1, f2);

  v2f Bop;
  Bop.x = hi ? f2 : f0;   // K2 rows live in lanes 16-31 of V0
  Bop.y = hi ? 0.0f : f1; // K3 pad / K1

  const size_t outbase = ((size_t)b * CH) * F_DIM + (size_t)fbase;

#pragma unroll
  for (int g = 0; g < 4; ++g) {
    const int row = g * 16 + m;
    v2f A;
    A.x = Wf[row * 4 + (hi ? 2 : 0)];
    A.y = Wf[row * 4 + (hi ? 3 : 1)];

    const int cbase = g * 16 + (hi ? 8 : 0);       // D rows M=v (lanes 0-15) / M=v+8
    v8f C;
#pragma unroll
    for (int v = 0; v < 8; ++v) C[v] = bf[cbase + v];

    // D = A(16x4 W') x B(4x16 feat) + C(bias)
    v8f D = __builtin_amdgcn_wmma_f32_16x16x4_f32(
        /*neg_a=*/false, A, /*neg_b=*/false, Bop,
        /*c_mod=*/(short)0, C, /*reuse_a=*/false, /*reuse_b=*/false);

#pragma unroll
    for (int v = 0; v < 8; ++v) {
      float val = fmaxf(D[v], 0.0f);
      // out[b][cbase+v][fbase+m]; never re-read -> non-temporal
      __builtin_nontemporal_store(val, &out[outbase + (size_t)(cbase + v) * F_DIM + m]);
    }
  }
}

extern "C" void kernel_launch(void* const* d_in, const int* in_sizes, int n_in,
                              void* d_out, int out_size, void* d_ws, size_t ws_size,
                              hipStream_t stream) {
  const float* normals = (const float*)d_in[0];  // [32,3,65536]
  const int*   ring    = (const int*)d_in[1];    // [32,65536,3]
  const float* conv_w  = (const float*)d_in[2];  // [64,3]
  const float* conv_b  = (const float*)d_in[3];  // [64]
  const float* gamma   = (const float*)d_in[4];  // [64]
  const float* beta    = (const float*)d_in[5];  // [64]
  float* out = (float*)d_out;                    // [32,64,65536]

  float* ws  = (float*)d_ws;
  float* mom = ws;            // 9 floats (aligned region of 64)
  float* Wf  = ws + 64;       // 64*4 folded weights
  float* bf  = ws + 64 + 256; // 64 folded biases

  hipMemsetAsync(mom, 0, 9 * sizeof(float), stream);

  moments_kernel<<<2048, 256, 0, stream>>>(normals, ring, mom);
  fold_kernel<<<1, 64, 0, stream>>>(conv_w, conv_b, gamma, beta, mom, Wf, bf);

  const int tiles  = B_DIM * (F_DIM / 16);       // 131072 wave-tiles
  const int blocks = tiles / 8;                  // 8 waves (256 thr) per block
  conv_bn_relu_kernel<<<blocks, 256, 0, stream>>>(normals, ring, Wf, bf, out);
}